// TriangleMultiplicationOutgoing_40956808135309
// MI455X (gfx1250) — compile-verified
//
#include <hip/hip_runtime.h>
#include <hip/hip_bf16.h>
#include <stdint.h>

// ---------------------------------------------------------------------------
// TriangleMultiplicationOutgoing on MI455X (gfx1250)
// B=1, N=512, CZ=C=128.  Memory-bound at fp32 compute rates, so we use the
// full-precision V_WMMA_F32_16X16X4_F32 path (D = A(16x4,f32) x B(4x16,f32) + C).
// ---------------------------------------------------------------------------

#define NN   512
#define CZ   128
#define CC   128
#define MM   (NN * NN)          // 262144 rows of z
#define EPS  1e-5f

typedef __attribute__((ext_vector_type(2))) float v2f;
typedef __attribute__((ext_vector_type(8))) float v8f;

__device__ __forceinline__ v8f wmma4(v2f a, v2f b, v8f c) {
  // 8-arg pattern: (neg_a, A, neg_b, B, c_mod, C, reuse_a, reuse_b)
  return __builtin_amdgcn_wmma_f32_16x16x4_f32(false, a, false, b, (short)0, c,
                                               false, false);
}

__device__ __forceinline__ float sigmoidf_fast(float x) {
  return 1.0f / (1.0f + __expf(-x));
}

// ---- CDNA5 async global->LDS copy (16B per lane), tracked by ASYNCcnt ------
__device__ __forceinline__ void async_copy16(uint32_t lds_off, const void* gaddr) {
  asm volatile("global_load_async_to_lds_b128 %0, %1, off"
               :: "v"(lds_off), "v"(gaddr) : "memory");
}
__device__ __forceinline__ void wait_async_le4() {
  asm volatile("s_wait_asynccnt 4" ::: "memory");
}
__device__ __forceinline__ void wait_async_le0() {
  asm volatile("s_wait_asynccnt 0" ::: "memory");
}

// ===========================================================================
// Kernel 1: fused LayerNorm(CZ) + 5 projections (w_gate_a, w_a, w_gate_b,
// w_b, w_gate) with sigmoid gating.  One block = 16 rows of z.
//   ach[c][m], bch[c][m]  (channel-major, m = i*N+k)  -- for the einsum
//   gout[m][c]            (raw gate logits, row-major)
// ===========================================================================
__global__ __launch_bounds__(256) void k1_ln_proj_gate(
    const float* __restrict__ x,
    const float* __restrict__ ln1g, const float* __restrict__ ln1b,
    const float* __restrict__ Wga,  const float* __restrict__ Wgb,
    const float* __restrict__ Wa,   const float* __restrict__ Wb,
    const float* __restrict__ Wg,
    float* __restrict__ ach, float* __restrict__ bch,
    float* __restrict__ gout)
{
  __shared__ float zt[16][129];     // normalized rows (pad 129: conflict-free)
  __shared__ float redS[16][16];
  __shared__ float redQ[16][16];
  __shared__ float muS[16], rsS[16];
  __shared__ float tt[128][17];     // transpose staging for channel-major out

  const int t  = threadIdx.x;
  const int m0 = blockIdx.x * 16;

  // ---------------- phase A: load 16x128 rows + LayerNorm -----------------
  {
    const int r  = t >> 4;          // row 0..15
    const int cg = t & 15;          // 8-col group
    float v[8];
    float s = 0.f, q = 0.f;
    const float* xr = x + (size_t)(m0 + r) * CZ + cg * 8;
#pragma unroll
    for (int i = 0; i < 8; ++i) { v[i] = xr[i]; s += v[i]; q += v[i] * v[i]; }
    redS[r][cg] = s; redQ[r][cg] = q;
    __syncthreads();
    if (cg == 0) {
      float a = 0.f, b = 0.f;
#pragma unroll
      for (int i = 0; i < 16; ++i) { a += redS[r][i]; b += redQ[r][i]; }
      float mu  = a * (1.0f / CZ);
      float var = b * (1.0f / CZ) - mu * mu;
      muS[r] = mu; rsS[r] = rsqrtf(var + EPS);
    }
    __syncthreads();
    const float mu = muS[r], rs = rsS[r];
#pragma unroll
    for (int i = 0; i < 8; ++i) {
      int c = cg * 8 + i;
      zt[r][c] = (v[i] - mu) * rs * ln1g[c] + ln1b[c];
    }
  }
  __syncthreads();

  // ---------------- phase B: 5 GEMM tiles per wave via WMMA ---------------
  const int lane = t & 31;
  const int w    = t >> 5;          // wave 0..7 -> 16-col block
  const int ml   = lane & 15;
  const int hi   = lane >> 4;
  const int ncol = w * 16 + ml;

  v8f cGA = {}, cGB = {}, cA = {}, cB = {}, cG = {};
  for (int kk = 0; kk < CZ; kk += 4) {
    const int ka = kk + hi * 2;
    v2f af; af.x = zt[ml][ka]; af.y = zt[ml][ka + 1];
    const int o0 = ka * CC + ncol;
    v2f bga; bga.x = Wga[o0]; bga.y = Wga[o0 + CC];
    v2f bgb; bgb.x = Wgb[o0]; bgb.y = Wgb[o0 + CC];
    v2f ba;  ba.x  = Wa[o0];  ba.y  = Wa[o0 + CC];
    v2f bb;  bb.x  = Wb[o0];  bb.y  = Wb[o0 + CC];
    v2f bg;  bg.x  = Wg[o0];  bg.y  = Wg[o0 + CC];
    cGA = wmma4(af, bga, cGA);
    cA  = wmma4(af, ba,  cA);
    cGB = wmma4(af, bgb, cGB);
    cB  = wmma4(af, bb,  cB);
    cG  = wmma4(af, bg,  cG);
  }

  // gate logits: row-major, coalesced
#pragma unroll
  for (int vI = 0; vI < 8; ++vI) {
    const int mloc = vI + hi * 8;
    gout[(size_t)(m0 + mloc) * CC + ncol] = cG[vI];
  }

  // a = sigmoid(zWga) * zWa  -> LDS transpose -> channel-major store
#pragma unroll
  for (int vI = 0; vI < 8; ++vI) {
    const int mloc = vI + hi * 8;
    tt[ncol][mloc] = cA[vI] * sigmoidf_fast(cGA[vI]);
  }
  __syncthreads();
  {
    const int c = t >> 1, half = t & 1;
    float* dst = ach + (size_t)c * MM + m0 + half * 8;
#pragma unroll
    for (int i = 0; i < 8; ++i) dst[i] = tt[c][half * 8 + i];
  }
  __syncthreads();

  // b = sigmoid(zWgb) * zWb
#pragma unroll
  for (int vI = 0; vI < 8; ++vI) {
    const int mloc = vI + hi * 8;
    tt[ncol][mloc] = cB[vI] * sigmoidf_fast(cGB[vI]);
  }
  __syncthreads();
  {
    const int c = t >> 1, half = t & 1;
    float* dst = bch + (size_t)c * MM + m0 + half * 8;
#pragma unroll
    for (int i = 0; i < 8; ++i) dst[i] = tt[c][half * 8 + i];
  }
}

// ===========================================================================
// Kernel 2: triangle einsum.  One block = (channel c, 64x64 (i,j) tile).
// upd_c = A_c * B_c^T, K = 512 in 16 tiles of 32, double-buffered async
// global->LDS staging (ASYNCcnt), WMMA f32 16x16x4 inner loop.
// Pad = 36 floats: 144B row stride (16B aligned for b128) and gcd-free banks.
// ===========================================================================
__global__ __launch_bounds__(256) void k2_triangle(
    const float* __restrict__ ach, const float* __restrict__ bch,
    float* __restrict__ updch)
{
  __shared__ float sA[2][64][36];
  __shared__ float sB[2][64][36];

  const int t  = threadIdx.x;
  const int bx = blockIdx.x;
  const int c  = bx >> 6;
  const int ti = (bx & 63) >> 3;
  const int tj = bx & 7;
  const int i0 = ti * 64;
  const int j0 = tj * 64;

  const float* Ag = ach + (size_t)c * MM + (size_t)i0 * NN;  // [64][512]
  const float* Bg = bch + (size_t)c * MM + (size_t)j0 * NN;  // [64][512]

  const int lane = t & 31;
  const int w    = t >> 5;
  const int ml   = lane & 15;
  const int hi   = lane >> 4;
  const int mi   = w & 3;              // row tile 0..3
  const int nj0  = (w >> 2) * 2;       // col tiles nj0, nj0+1

  // per-thread slot for staging: 512 float4 per matrix / 256 threads = 2
  const int row0 = (t) >> 3;
  const int col0 = ((t) & 7) * 4;
  const int row1 = (t + 256) >> 3;
  const int col1 = ((t + 256) & 7) * 4;

  auto stage = [&](int buf, int kk) {
    async_copy16((uint32_t)(uintptr_t)&sA[buf][row0][col0], Ag + (size_t)row0 * NN + kk + col0);
    async_copy16((uint32_t)(uintptr_t)&sB[buf][row0][col0], Bg + (size_t)row0 * NN + kk + col0);
    async_copy16((uint32_t)(uintptr_t)&sA[buf][row1][col1], Ag + (size_t)row1 * NN + kk + col1);
    async_copy16((uint32_t)(uintptr_t)&sB[buf][row1][col1], Bg + (size_t)row1 * NN + kk + col1);
  };

  v8f acc0 = {}, acc1 = {};

  stage(0, 0);
  for (int kt = 0; kt < 16; ++kt) {
    const int cur = kt & 1;
    __syncthreads();                       // everyone done reading buf cur^1
    if (kt + 1 < 16) { stage(cur ^ 1, (kt + 1) * 32); wait_async_le4(); }
    else             { wait_async_le0(); }
    __syncthreads();                       // buf cur fully staged

#pragma unroll
    for (int k4 = 0; k4 < 32; k4 += 4) {
      const int ka = k4 + hi * 2;
      v2f af;
      af.x = sA[cur][mi * 16 + ml][ka];
      af.y = sA[cur][mi * 16 + ml][ka + 1];
      v2f b0;
      b0.x = sB[cur][nj0 * 16 + ml][ka];
      b0.y = sB[cur][nj0 * 16 + ml][ka + 1];
      v2f b1;
      b1.x = sB[cur][(nj0 + 1) * 16 + ml][ka];
      b1.y = sB[cur][(nj0 + 1) * 16 + ml][ka + 1];
      acc0 = wmma4(af, b0, acc0);
      acc1 = wmma4(af, b1, acc1);
    }
  }

  const size_t base = (size_t)c * MM;
#pragma unroll
  for (int q = 0; q < 2; ++q) {
    v8f acc = q ? acc1 : acc0;
    const int j = j0 + (nj0 + q) * 16 + ml;
#pragma unroll
    for (int vI = 0; vI < 8; ++vI) {
      const int i = i0 + mi * 16 + vI + hi * 8;
      updch[base + (size_t)i * NN + j] = acc[vI];
    }
  }
}

// ===========================================================================
// Kernel 3: gather channel-major upd -> LDS transpose, LayerNorm(C),
// WMMA against w_out (C x CZ), multiply by sigmoid(gate), store out[m][cz].
// One block = 64 rows.
// ===========================================================================
__global__ __launch_bounds__(256) void k3_ln_out(
    const float* __restrict__ updch,
    const float* __restrict__ ln2g, const float* __restrict__ ln2b,
    const float* __restrict__ Wout,
    const float* __restrict__ gateRaw,
    float* __restrict__ out)
{
  __shared__ float u[64][129];

  const int t  = threadIdx.x;
  const int m0 = blockIdx.x * 64;

  // gather + transpose: 64 rows x 128 channels
#pragma unroll 4
  for (int it = 0; it < 32; ++it) {
    const int idx = it * 256 + t;
    const int cch = idx >> 6;
    const int mlc = idx & 63;
    u[mlc][cch] = updch[(size_t)cch * MM + m0 + mlc];
  }
  __syncthreads();

  if (t < 64) {
    float s = 0.f, q = 0.f;
    for (int c2 = 0; c2 < CC; ++c2) { float x = u[t][c2]; s += x; q += x * x; }
    const float mu  = s * (1.0f / CC);
    const float var = q * (1.0f / CC) - mu * mu;
    const float rs  = rsqrtf(var + EPS);
    for (int c2 = 0; c2 < CC; ++c2)
      u[t][c2] = (u[t][c2] - mu) * rs * ln2g[c2] + ln2b[c2];
  }
  __syncthreads();

  const int lane = t & 31;
  const int w    = t >> 5;
  const int ml   = lane & 15;
  const int hi   = lane >> 4;
  const int rt   = w & 3;              // row tile 0..3
  const int ct0  = (w >> 2) * 4;       // col tiles ct0..ct0+3

  v8f acc[4] = {};
  for (int kk = 0; kk < CC; kk += 4) {
    const int ka = kk + hi * 2;
    v2f af; af.x = u[rt * 16 + ml][ka]; af.y = u[rt * 16 + ml][ka + 1];
#pragma unroll
    for (int qd = 0; qd < 4; ++qd) {
      const int nc = (ct0 + qd) * 16 + ml;
      v2f bf; bf.x = Wout[ka * CZ + nc]; bf.y = Wout[(ka + 1) * CZ + nc];
      acc[qd] = wmma4(af, bf, acc[qd]);
    }
  }

#pragma unroll
  for (int qd = 0; qd < 4; ++qd) {
    const int nc = (ct0 + qd) * 16 + ml;
#pragma unroll
    for (int vI = 0; vI < 8; ++vI) {
      const int m = m0 + rt * 16 + vI + hi * 8;
      const float gg = sigmoidf_fast(gateRaw[(size_t)m * CZ + nc]);
      out[(size_t)m * CZ + nc] = acc[qd][vI] * gg;
    }
  }
}

// ===========================================================================
extern "C" void kernel_launch(void* const* d_in, const int* in_sizes, int n_in,
                              void* d_out, int out_size, void* d_ws, size_t ws_size,
                              hipStream_t stream) {
  const float* pair  = (const float*)d_in[0];
  const float* ln1g  = (const float*)d_in[1];
  const float* ln1b  = (const float*)d_in[2];
  const float* wga   = (const float*)d_in[3];
  const float* wgb   = (const float*)d_in[4];
  const float* wa    = (const float*)d_in[5];
  const float* wb    = (const float*)d_in[6];
  const float* ln2g  = (const float*)d_in[7];
  const float* ln2b  = (const float*)d_in[8];
  const float* wout  = (const float*)d_in[9];
  const float* wgate = (const float*)d_in[10];
  float* out = (float*)d_out;

  const size_t planeF = (size_t)CC * MM;   // 33,554,432 floats per buffer
  float* ach   = (float*)d_ws;
  float* bch   = ach + planeF;
  float* gws   = bch + planeF;
  float* updch = gws + planeF;

  k1_ln_proj_gate<<<MM / 16, 256, 0, stream>>>(
      pair, ln1g, ln1b, wga, wgb, wa, wb, wgate, ach, bch, gws);

  k2_triangle<<<CC * (NN / 64) * (NN / 64), 256, 0, stream>>>(ach, bch, updch);

  k3_ln_out<<<MM / 64, 256, 0, stream>>>(updch, ln2g, ln2b, wout, gws, out);
}